// PointNetLK_38689065402802
// MI455X (gfx1250) — compile-verified
//
#include <hip/hip_runtime.h>
#include <hip/hip_bf16.h>
#include <math.h>

#define BB    32
#define NN    1024
#define EMBD  1024
#define SPLIT 4
#define PTS_PER_WG (NN / SPLIT)   // 256
#define CHUNK 64
#define HSTR  66                  // LDS stride for 64-wide activations
#define HSTR2 130                 // LDS stride for 128-wide activations
#define XTOL_F 1e-7f

typedef float v2f __attribute__((ext_vector_type(2)));
typedef float v8f __attribute__((ext_vector_type(8)));

// ---------------------------------------------------------------------------
// WMMA helpers: D = relu(A * W^T + bias), fp32 16x16x4 tiles, wave32.
// A rows come from LDS, B (= W^T) comes from global (L2-resident weights).
// Lane layout (per CDNA5 ISA 7.12.2):
//   A 16x4 : lane l -> row l%16, v0 = A[row][k + 2*(l/16)], v1 = +1
//   B 4x16 : lane l -> col l%16, v0 = B[k + 2*(l/16)][col], v1 = +1
//   C/D    : lane l -> col l%16, VGPR v -> row v + 8*(l/16)
// ---------------------------------------------------------------------------
__device__ __forceinline__ void wmma_gemm_relu_store(
    const float* __restrict__ src, int sStr, int Kdim,
    const float* __restrict__ W, int ldw, const float* __restrict__ bias,
    float* __restrict__ dst, int dStr, int Nout,
    int Mbase, int lr, int lh)
{
  const int row = Mbase + lr;
  for (int nb = 0; nb < Nout; nb += 16) {
    float bv = bias[nb + lr];
    v8f c = {bv, bv, bv, bv, bv, bv, bv, bv};
#pragma unroll 4
    for (int kk = 0; kk < Kdim; kk += 4) {
      const int ka = kk + 2 * lh;
      v2f a  = *(const v2f*)(src + row * sStr + ka);
      v2f bm = *(const v2f*)(W + (size_t)(nb + lr) * ldw + ka);
      c = __builtin_amdgcn_wmma_f32_16x16x4_f32(false, a, false, bm,
                                                (short)0, c, false, false);
    }
#pragma unroll
    for (int v = 0; v < 8; ++v)
      dst[(Mbase + v + 8 * lh) * dStr + nb + lr] = fmaxf(c[v], 0.0f);
  }
}

__device__ __forceinline__ void wmma_gemm_relu_max(
    const float* __restrict__ src, int sStr, int Kdim,
    const float* __restrict__ W, int ldw, const float* __restrict__ bias,
    unsigned* __restrict__ mU, int Nout,
    int Mbase, int lr, int lh)
{
  const int row = Mbase + lr;
  for (int nb = 0; nb < Nout; nb += 16) {
    float bv = bias[nb + lr];
    v8f c = {bv, bv, bv, bv, bv, bv, bv, bv};
#pragma unroll 4
    for (int kk = 0; kk < Kdim; kk += 4) {
      const int ka = kk + 2 * lh;
      v2f a  = *(const v2f*)(src + row * sStr + ka);
      v2f bm = *(const v2f*)(W + (size_t)(nb + lr) * ldw + ka);
      c = __builtin_amdgcn_wmma_f32_16x16x4_f32(false, a, false, bm,
                                                (short)0, c, false, false);
    }
    float mx = c[0];
#pragma unroll
    for (int v = 1; v < 8; ++v) mx = fmaxf(mx, c[v]);
    mx = fmaxf(mx, 0.0f);   // relu; also makes bit-pattern max valid
    atomicMax(&mU[nb + lr], __float_as_uint(mx));
  }
}

// ---------------------------------------------------------------------------
// Fused PointNet: transform -> 5-layer MLP -> max-pool over points.
// Grid: (B*K*SPLIT) blocks of 128 threads (4 waves). Out must be pre-zeroed;
// results merged via global atomicMax on non-negative float bit patterns.
// ---------------------------------------------------------------------------
__global__ __launch_bounds__(128) void pointnet_kernel(
    const float* __restrict__ pts,     // B x N x 3 (centered)
    const float* __restrict__ T,       // transforms, 16 floats each (4x4 rm)
    int tIdxB, int tIdxK, int K,
    const float* __restrict__ W1, const float* __restrict__ B1,
    const float* __restrict__ W2, const float* __restrict__ B2,
    const float* __restrict__ W3, const float* __restrict__ B3,
    const float* __restrict__ W4, const float* __restrict__ B4,
    const float* __restrict__ W5, const float* __restrict__ B5,
    float* __restrict__ out)           // (B*K) x EMBD (uint-bit max merged)
{
  __shared__ __align__(16) float hA[CHUNK * HSTR];
  __shared__ __align__(16) float hB[CHUNK * HSTR2];
  __shared__ unsigned mU[EMBD];

  const int tid  = threadIdx.x;
  const int lane = tid & 31;
  const int wave = tid >> 5;
  const int lr   = lane & 15;
  const int lh   = lane >> 4;
  const int Mbase = wave * 16;

  const int blk = blockIdx.x;
  const int s   = blk % SPLIT;
  const int bk  = blk / SPLIT;
  const int b   = bk / K;
  const int k   = bk - b * K;

  const float* Tm = T + (size_t)(b * tIdxB + k * tIdxK) * 16;
  const float t00 = Tm[0], t01 = Tm[1], t02 = Tm[2],  t03 = Tm[3];
  const float t10 = Tm[4], t11 = Tm[5], t12 = Tm[6],  t13 = Tm[7];
  const float t20 = Tm[8], t21 = Tm[9], t22 = Tm[10], t23 = Tm[11];

  for (int i = tid; i < EMBD; i += 128) mU[i] = 0u;

  const float* P = pts + (size_t)b * NN * 3;
  const int pbase = s * PTS_PER_WG;

  for (int c0 = 0; c0 < PTS_PER_WG; c0 += CHUNK) {
    __syncthreads();                      // protect hA reuse across waves
    {
      // L1: 3 -> 64 (VALU; K too small for WMMA). thread t: point t%64,
      // channels [(t/64)*32, +32).
      const int p   = tid & 63;
      const int ch0 = (tid >> 6) * 32;
      const int gp  = pbase + c0 + p;
      const float x = P[gp * 3 + 0], y = P[gp * 3 + 1], z = P[gp * 3 + 2];
      const float xr = t00 * x + t01 * y + t02 * z + t03;
      const float yr = t10 * x + t11 * y + t12 * z + t13;
      const float zr = t20 * x + t21 * y + t22 * z + t23;
#pragma unroll
      for (int c = 0; c < 32; ++c) {
        const int ch = ch0 + c;
        float acc = W1[ch * 3 + 0] * xr + W1[ch * 3 + 1] * yr +
                    W1[ch * 3 + 2] * zr + B1[ch];
        hA[p * HSTR + ch] = fmaxf(acc, 0.0f);
      }
    }
    __syncthreads();
    // L2..L5: each wave owns rows [Mbase, Mbase+16) -> no cross-wave hazards
    wmma_gemm_relu_store(hA, HSTR,  64, W2, 64,  B2, hB, HSTR2, 64,  Mbase, lr, lh);
    wmma_gemm_relu_store(hB, HSTR2, 64, W3, 64,  B3, hA, HSTR,  64,  Mbase, lr, lh);
    wmma_gemm_relu_store(hA, HSTR,  64, W4, 64,  B4, hB, HSTR2, 128, Mbase, lr, lh);
    wmma_gemm_relu_max (hB, HSTR2, 128, W5, 128, B5, mU, EMBD,       Mbase, lr, lh);
  }
  __syncthreads();
  unsigned* og = (unsigned*)(out + (size_t)bk * EMBD);
  for (int i = tid; i < EMBD; i += 128) atomicMax(&og[i], mU[i]);
}

// ---------------------------------------------------------------------------
// Small support kernels
// ---------------------------------------------------------------------------
__global__ __launch_bounds__(256) void zero_kernel(float* p, int n) {
  int i = blockIdx.x * 256 + threadIdx.x;
  if (i < n) p[i] = 0.0f;
}

__global__ __launch_bounds__(256) void center_kernel(
    const float* __restrict__ tpl, const float* __restrict__ srcp,
    float* __restrict__ t0, float* __restrict__ s0,
    float* __restrict__ p0m, float* __restrict__ p1m)
{
  const int b = blockIdx.x, which = blockIdx.y, tid = threadIdx.x;
  const float* in = which ? srcp : tpl;
  float* outc = which ? s0 : t0;
  float* mean = which ? p1m : p0m;
  __shared__ float rx[256], ry[256], rz[256], m3[3];
  float sx = 0, sy = 0, sz = 0;
  for (int i = tid; i < NN; i += 256) {
    sx += in[((size_t)b * NN + i) * 3 + 0];
    sy += in[((size_t)b * NN + i) * 3 + 1];
    sz += in[((size_t)b * NN + i) * 3 + 2];
  }
  rx[tid] = sx; ry[tid] = sy; rz[tid] = sz;
  __syncthreads();
  for (int st = 128; st > 0; st >>= 1) {
    if (tid < st) { rx[tid] += rx[tid + st]; ry[tid] += ry[tid + st]; rz[tid] += rz[tid + st]; }
    __syncthreads();
  }
  if (tid == 0) {
    m3[0] = rx[0] / NN; m3[1] = ry[0] / NN; m3[2] = rz[0] / NN;
    mean[b * 3 + 0] = m3[0]; mean[b * 3 + 1] = m3[1]; mean[b * 3 + 2] = m3[2];
  }
  __syncthreads();
  for (int i = tid; i < NN; i += 256) {
    outc[((size_t)b * NN + i) * 3 + 0] = in[((size_t)b * NN + i) * 3 + 0] - m3[0];
    outc[((size_t)b * NN + i) * 3 + 1] = in[((size_t)b * NN + i) * 3 + 1] - m3[1];
    outc[((size_t)b * NN + i) * 3 + 2] = in[((size_t)b * NN + i) * 3 + 2] - m3[2];
  }
}

__device__ __forceinline__ void se3_exp_dev(const float x[6], float T[16]) {
  const float wx = x[0], wy = x[1], wz = x[2];
  const float vx = x[3], vy = x[4], vz = x[5];
  const float t = sqrtf(wx * wx + wy * wy + wz * wz + 1e-30f);
  const float tt = t * t;
  float s1, s2, s3;
  if (fabsf(t) < 0.01f) {
    s1 = 1.0f - tt / 6.0f * (1.0f - tt / 20.0f * (1.0f - tt / 42.0f));
    s2 = 0.5f * (1.0f - tt / 12.0f * (1.0f - tt / 30.0f * (1.0f - tt / 56.0f)));
    s3 = (1.0f / 6.0f) * (1.0f - tt / 20.0f * (1.0f - tt / 42.0f * (1.0f - tt / 72.0f)));
  } else {
    const float st = sinf(t), ct = cosf(t);
    s1 = st / t;
    s2 = (1.0f - ct) / tt;
    s3 = (t - st) / (tt * t);
  }
  const float W[9] = {0.f, -wz, wy, wz, 0.f, -wx, -wy, wx, 0.f};
  float S[9];
#pragma unroll
  for (int i = 0; i < 3; ++i)
#pragma unroll
    for (int j = 0; j < 3; ++j) {
      float a = 0.f;
#pragma unroll
      for (int q = 0; q < 3; ++q) a += W[i * 3 + q] * W[q * 3 + j];
      S[i * 3 + j] = a;
    }
  float V[9];
#pragma unroll
  for (int i = 0; i < 3; ++i)
#pragma unroll
    for (int j = 0; j < 3; ++j) {
      const float I_ = (i == j) ? 1.0f : 0.0f;
      T[i * 4 + j] = I_ + s1 * W[i * 3 + j] + s2 * S[i * 3 + j];
      V[i * 3 + j] = I_ + s2 * W[i * 3 + j] + s3 * S[i * 3 + j];
    }
  T[3]  = V[0] * vx + V[1] * vy + V[2] * vz;
  T[7]  = V[3] * vx + V[4] * vy + V[5] * vz;
  T[11] = V[6] * vx + V[7] * vy + V[8] * vz;
  T[12] = 0.f; T[13] = 0.f; T[14] = 0.f; T[15] = 1.f;
}

__global__ void setup_kernel(const float* __restrict__ dt,
                             float* idT, float* jacT, float* curT,
                             float* seriesRaw, unsigned* flagsU)
{
  const int tid = threadIdx.x;   // 32 threads
  if (tid < 6) {
    float x[6] = {0, 0, 0, 0, 0, 0};
    x[tid] = -dt[tid];
    se3_exp_dev(x, &jacT[tid * 16]);
  }
  if (tid == 6) {
#pragma unroll
    for (int i = 0; i < 16; ++i) idT[i] = (i % 5 == 0) ? 1.0f : 0.0f;
  }
  if (tid == 7) { flagsU[0] = 0u; flagsU[1] = 0u; }
  // b = tid: est_T0 = identity; series[0] = identity
  float* c = &curT[tid * 16];
#pragma unroll
  for (int i = 0; i < 16; ++i) {
    const float v = (i % 5 == 0) ? 1.0f : 0.0f;
    c[i] = v;
    seriesRaw[tid * 16 + i] = v;
  }
}

__device__ __forceinline__ void invert6(const float* A, float* Inv) {
  float M[6][12];
  for (int i = 0; i < 6; ++i)
    for (int j = 0; j < 6; ++j) {
      M[i][j] = A[i * 6 + j];
      M[i][6 + j] = (i == j) ? 1.0f : 0.0f;
    }
  for (int c = 0; c < 6; ++c) {
    int piv = c; float best = fabsf(M[c][c]);
    for (int r = c + 1; r < 6; ++r) {
      float v = fabsf(M[r][c]);
      if (v > best) { best = v; piv = r; }
    }
    if (piv != c)
      for (int j = 0; j < 12; ++j) { float t = M[c][j]; M[c][j] = M[piv][j]; M[piv][j] = t; }
    const float inv = 1.0f / M[c][c];
    for (int j = 0; j < 12; ++j) M[c][j] *= inv;
    for (int r = 0; r < 6; ++r)
      if (r != c) {
        const float f = M[r][c];
        for (int j = 0; j < 12; ++j) M[r][j] -= f * M[c][j];
      }
  }
  for (int i = 0; i < 6; ++i)
    for (int j = 0; j < 6; ++j) Inv[i * 6 + j] = M[i][6 + j];
}

__global__ __launch_bounds__(256) void pinv_kernel(
    const float* __restrict__ f0, const float* __restrict__ fk,
    const float* __restrict__ dt, float* __restrict__ pinv)
{
  const int b = blockIdx.x, tid = threadIdx.x;
  __shared__ float A[36], Inv[36];
  if (tid < 36) {
    const int i = tid / 6, j = tid % 6;
    float acc = 0.f;
    for (int m = 0; m < EMBD; ++m) {
      const float f0v = f0[(size_t)b * EMBD + m];
      const float Ji = (f0v - fk[((size_t)b * 6 + i) * EMBD + m]) / dt[i];
      const float Jj = (f0v - fk[((size_t)b * 6 + j) * EMBD + m]) / dt[j];
      acc += Ji * Jj;
    }
    A[tid] = acc;
  }
  __syncthreads();
  if (tid == 0) invert6(A, Inv);
  __syncthreads();
  for (int m = tid; m < EMBD; m += 256) {
    float Jrow[6];
    const float f0v = f0[(size_t)b * EMBD + m];
#pragma unroll
    for (int j = 0; j < 6; ++j)
      Jrow[j] = (f0v - fk[((size_t)b * 6 + j) * EMBD + m]) / dt[j];
#pragma unroll
    for (int i = 0; i < 6; ++i) {
      float acc = 0.f;
#pragma unroll
      for (int j = 0; j < 6; ++j) acc += Inv[i * 6 + j] * Jrow[j];
      pinv[((size_t)b * 6 + i) * EMBD + m] = acc;
    }
  }
}

__global__ __launch_bounds__(256) void pose_kernel(
    const float* __restrict__ f, const float* __restrict__ f0,
    const float* __restrict__ pinv, float* __restrict__ r,
    float* __restrict__ pose, unsigned* __restrict__ flagsU)
{
  const int b = blockIdx.x, tid = threadIdx.x;
  float acc[6] = {0, 0, 0, 0, 0, 0};
  for (int m = tid; m < EMBD; m += 256) {
    const float rv = f[(size_t)b * EMBD + m] - f0[(size_t)b * EMBD + m];
    r[(size_t)b * EMBD + m] = rv;
#pragma unroll
    for (int i = 0; i < 6; ++i) acc[i] += pinv[((size_t)b * 6 + i) * EMBD + m] * rv;
  }
  __shared__ float red[6][256];
#pragma unroll
  for (int i = 0; i < 6; ++i) red[i][tid] = acc[i];
  __syncthreads();
  for (int st = 128; st > 0; st >>= 1) {
    if (tid < st)
#pragma unroll
      for (int i = 0; i < 6; ++i) red[i][tid] += red[i][tid + st];
    __syncthreads();
  }
  if (tid == 0) {
    float n2 = 0.f;
#pragma unroll
    for (int i = 0; i < 6; ++i) {
      const float p = -red[i][0];
      pose[b * 6 + i] = p;
      n2 += p * p;
    }
    atomicMax(&flagsU[0], __float_as_uint(sqrtf(n2)));
  }
}

__global__ void update_kernel(unsigned* __restrict__ flagsU,
                              const float* __restrict__ pose,
                              float* __restrict__ curT,
                              float* __restrict__ seriesRaw, int it)
{
  __shared__ int sstop;
  const int tid = threadIdx.x;   // 32 threads, b = tid
  if (tid == 0) {
    const float check = __uint_as_float(flagsU[0]);
    unsigned stop = flagsU[1];
    if (check < XTOL_F) stop = 1u;
    flagsU[1] = stop;
    flagsU[0] = 0u;              // reset accumulator for next iteration
    sstop = (int)stop;
  }
  __syncthreads();
  float* Tc = &curT[tid * 16];
  if (!sstop) {
    float x[6];
#pragma unroll
    for (int i = 0; i < 6; ++i) x[i] = pose[tid * 6 + i];
    float G[16], NT[16];
    se3_exp_dev(x, G);
#pragma unroll
    for (int i = 0; i < 4; ++i)
#pragma unroll
      for (int j = 0; j < 4; ++j) {
        float a = 0.f;
#pragma unroll
        for (int q = 0; q < 4; ++q) a += G[i * 4 + q] * Tc[q * 4 + j];
        NT[i * 4 + j] = a;
      }
#pragma unroll
    for (int i = 0; i < 16; ++i) Tc[i] = NT[i];
  }
  float* S = &seriesRaw[(size_t)(it + 1) * BB * 16 + tid * 16];
#pragma unroll
  for (int i = 0; i < 16; ++i) S[i] = Tc[i];
}

__global__ __launch_bounds__(256) void final_kernel(
    const float* __restrict__ curT, const float* __restrict__ seriesRaw,
    const float* __restrict__ r, const float* __restrict__ p0m,
    const float* __restrict__ p1m, float* __restrict__ outEst,
    float* __restrict__ outR, float* __restrict__ outSeries)
{
  const int b = blockIdx.x, tid = threadIdx.x;
  for (int m = tid; m < EMBD; m += 256)
    outR[(size_t)b * EMBD + m] = r[(size_t)b * EMBD + m];
  if (tid < 12) {
    const float* M = (tid == 0) ? &curT[b * 16]
                                : &seriesRaw[(size_t)(tid - 1) * BB * 16 + b * 16];
    const float px = p0m[b * 3 + 0], py = p0m[b * 3 + 1], pz = p0m[b * 3 + 2];
    const float qx = p1m[b * 3 + 0], qy = p1m[b * 3 + 1], qz = p1m[b * 3 + 2];
    float A[16];
#pragma unroll
    for (int j = 0; j < 4; ++j) {
      A[0 * 4 + j] = M[0 * 4 + j] + px * M[3 * 4 + j];
      A[1 * 4 + j] = M[1 * 4 + j] + py * M[3 * 4 + j];
      A[2 * 4 + j] = M[2 * 4 + j] + pz * M[3 * 4 + j];
      A[3 * 4 + j] = M[3 * 4 + j];
    }
    float R[16];
#pragma unroll
    for (int i = 0; i < 4; ++i) {
#pragma unroll
      for (int j = 0; j < 3; ++j) R[i * 4 + j] = A[i * 4 + j];
      R[i * 4 + 3] = A[i * 4 + 3] - (A[i * 4 + 0] * qx + A[i * 4 + 1] * qy + A[i * 4 + 2] * qz);
    }
    float* dst = (tid == 0) ? &outEst[b * 16]
                            : &outSeries[((size_t)(tid - 1) * BB + b) * 16];
#pragma unroll
    for (int i = 0; i < 16; ++i) dst[i] = R[i];
  }
}

// ---------------------------------------------------------------------------
extern "C" void kernel_launch(void* const* d_in, const int* in_sizes, int n_in,
                              void* d_out, int out_size, void* d_ws, size_t ws_size,
                              hipStream_t stream) {
  (void)in_sizes; (void)n_in; (void)out_size; (void)ws_size;
  const float* tpl  = (const float*)d_in[0];
  const float* srcp = (const float*)d_in[1];
  const float* W1 = (const float*)d_in[2];  const float* B1 = (const float*)d_in[3];
  const float* W2 = (const float*)d_in[4];  const float* B2 = (const float*)d_in[5];
  const float* W3 = (const float*)d_in[6];  const float* B3 = (const float*)d_in[7];
  const float* W4 = (const float*)d_in[8];  const float* B4 = (const float*)d_in[9];
  const float* W5 = (const float*)d_in[10]; const float* B5 = (const float*)d_in[11];
  const float* dt = (const float*)d_in[12];
  float* out = (float*)d_out;
  float* ws  = (float*)d_ws;

  // workspace carve-up (floats)
  float* t0   = ws;                       // 32*1024*3
  float* s0   = t0 + BB * NN * 3;         // 32*1024*3
  float* p0m  = s0 + BB * NN * 3;         // 96
  float* p1m  = p0m + 96;                 // 96
  float* idT  = p1m + 96;                 // 16
  float* jacT = idT + 16;                 // 96
  float* curT = jacT + 96;                // 512
  float* f0   = curT + 512;               // 32768
  float* fk   = f0 + BB * EMBD;           // 196608  (contiguous with f0)
  float* pinvB= fk + BB * 6 * EMBD;       // 196608
  float* fcur = pinvB + BB * 6 * EMBD;    // 32768
  float* rbuf = fcur + BB * EMBD;         // 32768
  float* poseB= rbuf + BB * EMBD;         // 192
  unsigned* flagsU = (unsigned*)(poseB + 192);  // 4 slots
  float* seriesRaw = poseB + 192 + 4;     // 11*32*16 = 5632

  // 1) center clouds + means
  center_kernel<<<dim3(BB, 2), 256, 0, stream>>>(tpl, srcp, t0, s0, p0m, p1m);
  // 2) twists/identity/curT/flags
  setup_kernel<<<1, 32, 0, stream>>>(dt, idT, jacT, curT, seriesRaw, flagsU);
  // 3) f0 and Jacobian features (zero first: merged via atomicMax on bits)
  {
    const int nz = BB * EMBD + BB * 6 * EMBD;  // f0 + fk contiguous
    zero_kernel<<<(nz + 255) / 256, 256, 0, stream>>>(f0, nz);
  }
  pointnet_kernel<<<BB * 1 * SPLIT, 128, 0, stream>>>(
      t0, idT, 0, 0, 1, W1, B1, W2, B2, W3, B3, W4, B4, W5, B5, f0);
  pointnet_kernel<<<BB * 6 * SPLIT, 128, 0, stream>>>(
      t0, jacT, 0, 1, 6, W1, B1, W2, B2, W3, B3, W4, B4, W5, B5, fk);
  // 4) pinv = inv(J^T J) J^T
  pinv_kernel<<<BB, 256, 0, stream>>>(f0, fk, dt, pinvB);
  // 5) 10 LK iterations (maxiter = 10 per reference setup)
  for (int it = 0; it < 10; ++it) {
    zero_kernel<<<(BB * EMBD + 255) / 256, 256, 0, stream>>>(fcur, BB * EMBD);
    pointnet_kernel<<<BB * 1 * SPLIT, 128, 0, stream>>>(
        s0, curT, 1, 0, 1, W1, B1, W2, B2, W3, B3, W4, B4, W5, B5, fcur);
    pose_kernel<<<BB, 256, 0, stream>>>(fcur, f0, pinvB, rbuf, poseB, flagsU);
    update_kernel<<<1, 32, 0, stream>>>(flagsU, poseB, curT, seriesRaw, it);
  }
  // 6) compose with a0/a1 and emit outputs: est (512) | r_last (32768) | series (5632)
  final_kernel<<<BB, 256, 0, stream>>>(curT, seriesRaw, rbuf, p0m, p1m,
                                       out, out + 512, out + 512 + BB * EMBD);
}